// LightGCNWithContent_44263932952646
// MI455X (gfx1250) — compile-verified
//
#include <hip/hip_runtime.h>
#include <hip/hip_bf16.h>
#include <math.h>

typedef __attribute__((ext_vector_type(2))) float v2f;
typedef __attribute__((ext_vector_type(8))) float v8f;

#define EMBED_DIM   128
#define CONTENT_DIM 768
#define TILE_K      32

// ---------------------------------------------------------------- utilities
__global__ void zero_f_kernel(float* __restrict__ p, int n) {
    int i = blockIdx.x * 256 + threadIdx.x;
    if (i < n) p[i] = 0.0f;
}

// deg[dst] += ew  (in-degree weighted)
__global__ void deg_kernel(const int* __restrict__ dst, const float* __restrict__ ew,
                           float* __restrict__ deg, int E) {
    int e = blockIdx.x * 256 + threadIdx.x;
    if (e < E) atomicAdd(&deg[dst[e]], ew[e]);
}

// deg -> dinv (in place)
__global__ void dinv_kernel(float* __restrict__ deg, int N) {
    int i = blockIdx.x * 256 + threadIdx.x;
    if (i < N) {
        float v = deg[i];
        deg[i] = (v > 0.0f) ? rsqrtf(v) : 0.0f;
    }
}

// norm[e] = dinv[src]*ew*dinv[dst]
__global__ void norm_kernel(const int* __restrict__ src, const int* __restrict__ dst,
                            const float* __restrict__ ew, const float* __restrict__ dinv,
                            float* __restrict__ norm, int E) {
    int e = blockIdx.x * 256 + threadIdx.x;
    if (e < E) norm[e] = dinv[src[e]] * ew[e] * dinv[dst[e]];
}

// x_cur = concat(user_emb, item_emb); acc = same (acc is d_out+1: 4B aligned -> scalar)
__global__ void init_kernel(const float* __restrict__ ue, const float* __restrict__ ie,
                            float* __restrict__ xcur, float* __restrict__ acc,
                            int u128, int total) {
    int i = blockIdx.x * 256 + threadIdx.x;
    if (i >= total) return;
    float v = (i < u128) ? ue[i] : ie[i - u128];
    xcur[i] = v;
    acc[i]  = v;
}

// one wave per edge: x_next[dst,:] += norm * x_cur[src,:]
__global__ __launch_bounds__(256)
void scatter_kernel(const float* __restrict__ xcur, float* __restrict__ xnext,
                    const int* __restrict__ src, const int* __restrict__ dst,
                    const float* __restrict__ norm, int E) {
    int e = blockIdx.x * 8 + (threadIdx.x >> 5);
    if (e >= E) return;
    int lane = threadIdx.x & 31;
    float w = norm[e];
    int s = src[e], d = dst[e];
    float4 v = ((const float4*)(xcur + (size_t)s * EMBED_DIM))[lane];
    float* o = xnext + (size_t)d * EMBED_DIM + lane * 4;
    atomicAdd(o + 0, w * v.x);
    atomicAdd(o + 1, w * v.y);
    atomicAdd(o + 2, w * v.z);
    atomicAdd(o + 3, w * v.w);
}

// acc += xn (acc misaligned for vec -> scalar)
__global__ void add_kernel(float* __restrict__ acc, const float* __restrict__ xn, int n) {
    int i = blockIdx.x * 256 + threadIdx.x;
    if (i < n) acc[i] += xn[i];
}

__global__ void scale_kernel(float* __restrict__ acc, float s, int n) {
    int i = blockIdx.x * 256 + threadIdx.x;
    if (i < n) acc[i] *= s;
}

// ------------------------------------------------- BPR + reg (wave per sample)
__global__ __launch_bounds__(256)
void bpr_kernel(const float* __restrict__ xf, const int* __restrict__ bu,
                const int* __restrict__ bp, const int* __restrict__ bn,
                float* __restrict__ sums, int B, int U) {
    int b = blockIdx.x * 8 + (threadIdx.x >> 5);
    if (b >= B) return;
    int lane = threadIdx.x & 31;
    const float* u = xf + (size_t)bu[b] * EMBED_DIM;
    const float* p = xf + (size_t)(U + bp[b]) * EMBED_DIM;
    const float* n = xf + (size_t)(U + bn[b]) * EMBED_DIM;
    float pos = 0.f, neg = 0.f, sq = 0.f;
    #pragma unroll
    for (int j = 0; j < 4; ++j) {
        int d = lane + j * 32;
        float uv = u[d], pv = p[d], nv = n[d];
        pos += uv * pv;
        neg += uv * nv;
        sq  += uv * uv + pv * pv + nv * nv;
    }
    #pragma unroll
    for (int off = 16; off; off >>= 1) {
        pos += __shfl_xor(pos, off, 32);
        neg += __shfl_xor(neg, off, 32);
        sq  += __shfl_xor(sq,  off, 32);
    }
    if (lane == 0) {
        float x  = neg - pos;                                // -(pos-neg)
        float sp = fmaxf(x, 0.f) + log1pf(expf(-fabsf(x)));  // stable softplus
        atomicAdd(&sums[0], sp);
        atomicAdd(&sums[1], sq);
    }
}

// ---------------------- content alignment: WMMA f32 GEMM + fused sq-diff reduce
// proj = content[M x 768] @ W^T (W is [128 x 768]); align += (ifinal - proj)^2
// Tiles staged with GLOBAL_LOAD_ASYNC_TO_LDS_B128 (ASYNCcnt path, no VGPR bounce).
// Dynamic LDS so raw byte offsets start at 0: As @ [0,16384), Ws @ [16384,32768).
__global__ __launch_bounds__(256)
void align_wmma_kernel(const float* __restrict__ content, const float* __restrict__ W,
                       const float* __restrict__ ifinal, float* __restrict__ sums, int M) {
    extern __shared__ float smem[];
    float* As = smem;                  // 128*TILE_K floats, LDS byte offset 0
    float* Ws = smem + 128 * TILE_K;   // LDS byte offset 16384

    int tid  = threadIdx.x;
    int lane = tid & 31, wv = tid >> 5;
    int l16  = lane & 15, half = lane >> 4;
    int row_base = blockIdx.x * 128;

    v8f c[8];
    v8f zero = {};
    #pragma unroll
    for (int t = 0; t < 8; ++t) c[t] = zero;

    for (int kc = 0; kc < CONTENT_DIM; kc += TILE_K) {
        // async fill: 128 rows x 32 k = 1024 float4 slots per matrix, 8 copies/thread
        #pragma unroll
        for (int i = 0; i < 4; ++i) {
            int s   = tid + i * 256;
            int row = s >> 3;      // 0..127
            int c4  = s & 7;       // 0..7 (float4 column)
            int ar  = row_base + row;
            if (ar >= M) ar = M - 1;   // clamp (masked in reduction)
            unsigned ldsA = (unsigned)((row * TILE_K + c4 * 4) * 4);
            const float* gA = content + (size_t)ar * CONTENT_DIM + kc + c4 * 4;
            asm volatile("global_load_async_to_lds_b128 %0, %1, off"
                         :: "v"(ldsA), "v"(gA) : "memory");
            unsigned ldsW = (unsigned)(128 * TILE_K * 4 + (row * TILE_K + c4 * 4) * 4);
            const float* gW = W + (size_t)row * CONTENT_DIM + kc + c4 * 4;
            asm volatile("global_load_async_to_lds_b128 %0, %1, off"
                         :: "v"(ldsW), "v"(gW) : "memory");
        }
        asm volatile("s_wait_asynccnt 0x0" ::: "memory");
        __syncthreads();

        int arow = wv * 16 + l16;
        #pragma unroll
        for (int k4 = 0; k4 < TILE_K; k4 += 4) {
            int ko = k4 + half * 2;                         // even -> 8B aligned
            v2f a = *(const v2f*)(As + arow * TILE_K + ko); // A 16x4 per-lane layout
            #pragma unroll
            for (int t = 0; t < 8; ++t) {
                v2f b = *(const v2f*)(Ws + (t * 16 + l16) * TILE_K + ko); // B 4x16
                c[t] = __builtin_amdgcn_wmma_f32_16x16x4_f32(
                    /*neg_a=*/false, a, /*neg_b=*/false, b,
                    /*c_mod=*/(short)0, c[t], /*reuse_a=*/false, /*reuse_b=*/false);
            }
        }
        __syncthreads();
    }

    // epilogue: squared error vs i_final, masked on valid rows
    float local = 0.f;
    #pragma unroll
    for (int t = 0; t < 8; ++t) {
        int ncol = t * 16 + l16;
        #pragma unroll
        for (int r = 0; r < 8; ++r) {
            int grow = row_base + wv * 16 + half * 8 + r;   // C-layout row for VGPR r
            if (grow < M) {
                float d = c[t][r] - ifinal[(size_t)grow * EMBED_DIM + ncol];
                local += d * d;
            }
        }
    }
    #pragma unroll
    for (int off = 16; off; off >>= 1) local += __shfl_xor(local, off, 32);
    if (lane == 0) atomicAdd(&sums[2], local);
}

__global__ void finalize_kernel(const float* __restrict__ sums, float* __restrict__ out,
                                float inv_batch, float inv_align) {
    float bpr   = sums[0] * inv_batch;
    float reg   = sums[1] * inv_batch * 1e-4f;
    float align = sums[2] * inv_align;
    out[0] = bpr + reg + 0.1f * align;
}

// ---------------------------------------------------------------- launcher
extern "C" void kernel_launch(void* const* d_in, const int* in_sizes, int n_in,
                              void* d_out, int out_size, void* d_ws, size_t ws_size,
                              hipStream_t stream) {
    const int*   edge_index = (const int*)  d_in[0];
    const float* edge_w     = (const float*)d_in[1];
    const int*   b_users    = (const int*)  d_in[2];
    const int*   b_pos      = (const int*)  d_in[3];
    const int*   b_neg      = (const int*)  d_in[4];
    const float* content    = (const float*)d_in[5];
    const float* user_emb   = (const float*)d_in[6];
    const float* item_emb   = (const float*)d_in[7];
    const float* proj_w     = (const float*)d_in[8];

    const int E = in_sizes[1];
    const int B = in_sizes[2];
    const int U = in_sizes[6] / EMBED_DIM;
    const int I = in_sizes[7] / EMBED_DIM;
    const int N = U + I;
    const int ND = N * EMBED_DIM;           // 19.2M floats

    const int* src = edge_index;            // edge_index[0,:]
    const int* dst = edge_index + E;        // edge_index[1,:]

    // workspace layout (floats)
    float* ws   = (float*)d_ws;
    float* deg  = ws;                       // N (becomes dinv in place)
    float* nrm  = deg + N;                  // E
    float* xa   = nrm + E;                  // N*128 (16B aligned: (N+E)*4 % 16 == 0)
    float* xb   = xa + ND;                  // N*128
    float* sums = xb + ND;                  // 3

    float* outv = (float*)d_out;
    float* acc  = outv + 1;                 // u_final ‖ i_final lives here

    const int ZB = 256;
    auto blocks = [](int n, int per) { return (unsigned)((n + per - 1) / per); };

    // degrees -> dinv -> per-edge norms
    zero_f_kernel<<<blocks(N, ZB), ZB, 0, stream>>>(deg, N);
    zero_f_kernel<<<1, 32, 0, stream>>>(sums, 3);
    deg_kernel <<<blocks(E, ZB), ZB, 0, stream>>>(dst, edge_w, deg, E);
    dinv_kernel<<<blocks(N, ZB), ZB, 0, stream>>>(deg, N);
    norm_kernel<<<blocks(E, ZB), ZB, 0, stream>>>(src, dst, edge_w, deg, nrm, E);

    // x0 and acc
    init_kernel<<<blocks(ND, ZB), ZB, 0, stream>>>(user_emb, item_emb, xa, acc,
                                                   U * EMBED_DIM, ND);

    // 3 LGConv layers with ping-pong
    float* xcur = xa;
    float* xnext = xb;
    for (int layer = 0; layer < 3; ++layer) {
        zero_f_kernel<<<blocks(ND, ZB), ZB, 0, stream>>>(xnext, ND);
        scatter_kernel<<<blocks(E, 8), 256, 0, stream>>>(xcur, xnext, src, dst, nrm, E);
        add_kernel<<<blocks(ND, ZB), ZB, 0, stream>>>(acc, xnext, ND);
        float* t = xcur; xcur = xnext; xnext = t;
    }
    scale_kernel<<<blocks(ND, ZB), ZB, 0, stream>>>(acc, 0.25f, ND);

    // losses
    bpr_kernel<<<blocks(B, 8), 256, 0, stream>>>(acc, b_users, b_pos, b_neg, sums, B, U);
    const float* ifinal = acc + (size_t)U * EMBED_DIM;
    align_wmma_kernel<<<blocks(I, 128), 256, 32768, stream>>>(content, proj_w, ifinal,
                                                              sums, I);

    finalize_kernel<<<1, 1, 0, stream>>>(sums, outv, 1.0f / (float)B,
                                         1.0f / ((float)I * (float)EMBED_DIM));
}